// VehCollLoss_9998683865517
// MI455X (gfx1250) — compile-verified
//
#include <hip/hip_runtime.h>

typedef __attribute__((ext_vector_type(2))) float v2f;
typedef __attribute__((ext_vector_type(8))) float v8f;

constexpr int kNA = 192;
constexpr int kT  = 80;
constexpr int kNC = 5;

// One wave32 per (t, rowTile, colTile). Squared distances via
// V_WMMA_F32_16X16X4_F32: d2 = [x,y,1,s] . [-2x', -2y', s', 1]^T  (K=4).
__global__ __launch_bounds__(256) void veh_coll_wmma_kernel(
    const float* __restrict__ traj,    // (NA, T, 4): x, y, hx, hy
    const float* __restrict__ cent,    // (NA, 5, 4): cx, cy, 1, 0
    const float* __restrict__ pdist,   // (NA, NA)
    float* __restrict__ out)           // penalties (T,NA,NA) then mask (T,NA,NA)
{
  const int lane = threadIdx.x & 31;
  const int hf   = lane >> 4;   // which half of the wave (K-split for A/B operands)
  const int l16  = lane & 15;

  const int wid = (blockIdx.x * blockDim.x + threadIdx.x) >> 5;
  const int tj = wid % 12;
  const int ti = (wid / 12) % 12;
  const int t  = wid / 144;

  const int i0 = ti * 16;
  const int j0 = tj * 16;

  // Both lane halves compute the same agent's data (M index = l16), so the
  // A/B VGPR layouts (K={0,2} in v0, K={1,3} in v1, split by lane half) need
  // no cross-lane communication.
  const int ra = i0 + l16;   // row agent
  const int ca = j0 + l16;   // col agent

  const float4 trj = ((const float4*)traj)[ra * kT + t];
  const float4 tcj = ((const float4*)traj)[ca * kT + t];

  const float rinv = 1.0f / sqrtf(trj.z * trj.z + trj.w * trj.w);
  const float rhx = trj.z * rinv, rhy = trj.w * rinv;
  const float cinv = 1.0f / sqrtf(tcj.z * tcj.z + tcj.w * tcj.w);
  const float chx = tcj.z * cinv, chy = tcj.w * cinv;

  float rwx[kNC], rwy[kNC], rss[kNC];
  float cwx[kNC], cwy[kNC], css[kNC];
#pragma unroll
  for (int c = 0; c < kNC; ++c) {
    const float4 ce = ((const float4*)cent)[ra * kNC + c];
    const float wx = trj.x + rhx * ce.x - rhy * ce.y;
    const float wy = trj.y + rhy * ce.x + rhx * ce.y;
    rwx[c] = wx; rwy[c] = wy; rss[c] = wx * wx + wy * wy;

    const float4 cd = ((const float4*)cent)[ca * kNC + c];
    const float vx = tcj.x + chx * cd.x - chy * cd.y;
    const float vy = tcj.y + chy * cd.x + chx * cd.y;
    cwx[c] = vx; cwy[c] = vy; css[c] = vx * vx + vy * vy;
  }

  v8f mind2;
#pragma unroll
  for (int r = 0; r < 8; ++r) mind2[r] = 3.0e38f;

  // 25 circle-pair WMMAs; elementwise min-fold of the 16x16 d2 tiles.
#pragma unroll
  for (int c = 0; c < kNC; ++c) {
    v2f a;
    a.x = hf ? 1.0f   : rwx[c];   // K=0: x      | K=2: 1
    a.y = hf ? rss[c] : rwy[c];   // K=1: y      | K=3: |p|^2
#pragma unroll
    for (int d = 0; d < kNC; ++d) {
      v2f b;
      b.x = hf ? css[d] : -2.0f * cwx[d];  // K=0: -2x' | K=2: |q|^2
      b.y = hf ? 1.0f   : -2.0f * cwy[d];  // K=1: -2y' | K=3: 1
      v8f z = {};
      v8f d2 = __builtin_amdgcn_wmma_f32_16x16x4_f32(
          /*neg_a=*/false, a, /*neg_b=*/false, b,
          /*c_mod=*/(short)0, z, /*reuse_a=*/false, /*reuse_b=*/false);
#pragma unroll
      for (int r = 0; r < 8; ++r) mind2[r] = fminf(mind2[r], d2[r]);
    }
  }

  float* __restrict__ pen = out;
  float* __restrict__ msk = out + (size_t)kT * kNA * kNA;
  const size_t tb = (size_t)t * kNA * kNA;

#pragma unroll
  for (int r = 0; r < 8; ++r) {
    const int arow = i0 + r + 8 * hf;       // C/D layout: VGPR r, half -> M = r + 8*half
    const int acol = j0 + l16;              // N = lane % 16
    const float pd = pdist[arow * kNA + acol];
    const float dmin = sqrtf(fmaxf(mind2[r], 0.0f));
    const size_t idx = tb + (size_t)arow * kNA + acol;
    pen[idx] = 1.0f - dmin / pd;
    msk[idx] = (dmin <= pd && arow != acol) ? 1.0f : 0.0f;
  }
}

extern "C" void kernel_launch(void* const* d_in, const int* in_sizes, int n_in,
                              void* d_out, int out_size, void* d_ws, size_t ws_size,
                              hipStream_t stream) {
  (void)in_sizes; (void)n_in; (void)d_ws; (void)ws_size; (void)out_size;
  const float* traj  = (const float*)d_in[0];   // (192, 80, 4) f32
  const float* cent  = (const float*)d_in[1];   // (192, 5, 4)  f32
  const float* pdist = (const float*)d_in[2];   // (192, 192)   f32
  // d_in[3] (off_diag_mask) is just (i != j); computed in-kernel.

  // 80 * 12 * 12 = 11520 waves; 8 waves per 256-thread block -> 1440 blocks.
  const int waves  = kT * (kNA / 16) * (kNA / 16);
  const int blocks = (waves * 32) / 256;
  veh_coll_wmma_kernel<<<blocks, 256, 0, stream>>>(traj, cent, pdist, (float*)d_out);
}